// FullyConnectedAtomicSpace_64759516889241
// MI455X (gfx1250) — compile-verified
//
#include <hip/hip_runtime.h>
#include <hip/hip_bf16.h>
#include <stdint.h>

// ---------------------------------------------------------------------------
// Exact rank-counting via two-level histogram (top-17 bits in L2, low-15 bits
// in 128KB of CDNA5 LDS), WMMA-accelerated prefix scan, TDM-staged bins.
//
// Roofline: this is a sort/rank problem, not a matmul; total HBM traffic is
// ~650MB -> ~30-50us at 23.3 TB/s. CDNA5 features used: 128KB LDS histogram
// per workgroup (impossible on 64KB CDNA4 CUs), v_wmma_f32_16x16x4_f32 for an
// exact integer prefix scan (all counts <= 2^24 are exact in f32), and the
// Tensor Data Mover (tensor_load_to_lds + s_wait_tensorcnt) to stage bin keys.
// ---------------------------------------------------------------------------

static constexpr uint32_t Hdim   = 4096;
static constexpr uint32_t Wdim   = 4096;
static constexpr uint32_t NPix   = Hdim * Wdim;      // 2^24 — all counts exact in f32
static constexpr uint32_t NBins  = 1u << 17;         // level-1: top 17 key bits
static constexpr uint32_t L2Num  = 1u << 15;         // level-2: low 15 key bits
static constexpr uint32_t StageCap = 2048;           // TDM-staged bin size cap
static constexpr uint32_t RankSmemBytes = (L2Num + StageCap + 256) * sizeof(uint32_t);

typedef float v2f __attribute__((ext_vector_type(2)));
typedef float v8f __attribute__((ext_vector_type(8)));
typedef unsigned int v4u __attribute__((ext_vector_type(4)));
typedef int  v4i __attribute__((ext_vector_type(4)));
typedef int  v8i __attribute__((ext_vector_type(8)));

// Monotonic float -> u32 key (order-preserving)
__device__ __forceinline__ uint32_t fkey(float f) {
  uint32_t u = __float_as_uint(f);
  return (u & 0x80000000u) ? ~u : (u | 0x80000000u);
}

// -------------------------- K0: zero counters ------------------------------
__global__ __launch_bounds__(256) void zero_kernel(uint32_t* __restrict__ p, uint32_t n) {
  uint32_t i = blockIdx.x * 256 + threadIdx.x;
  if (i < n) p[i] = 0u;
}

// -------------------------- K1: level-1 histogram --------------------------
__global__ __launch_bounds__(256) void hist_kernel(const float* __restrict__ ch,
                                                   uint32_t* __restrict__ hist) {
  uint32_t i = blockIdx.x * 256 + threadIdx.x;
  __builtin_prefetch(ch + i + 16384, 0, 0);        // -> global_prefetch_b8
  atomicAdd(&hist[fkey(ch[i]) >> 15], 1u);         // 512KB of counters: stays in L2
}

// --------------- K2: exclusive scan of 131072 bins (1 block) ---------------
// 1024 threads * 128-chunk sequential scans; the 16 group sums are
// prefix-summed with chained V_WMMA_F32_16X16X4_F32 (exact: integers <= 2^24).
// A-operand layout per ISA 7.12.2 (16x4 f32): lane l holds M=l&15,
// K = 2*(l>>4) + vgpr. B = all-ones (layout-invariant).
__global__ __launch_bounds__(1024) void scan_kernel(uint32_t* __restrict__ hist) {
  __shared__ uint32_t part[1024];
  __shared__ float gsum[16];
  __shared__ float goff[16];
  const int t = threadIdx.x;
  const int base = t * 128;

  uint32_t s = 0;
#pragma unroll 4
  for (int i = 0; i < 128; ++i) s += hist[base + i];
  part[t] = s;
  __syncthreads();

  if (t < 16) {                       // 16 groups of 64 partials
    uint32_t g = 0;
    for (int j = 0; j < 64; ++j) g += part[t * 64 + j];
    gsum[t] = (float)g;
  }
  __syncthreads();

  if (t < 32) {                       // wave 0 (EXEC all-ones): WMMA exclusive scan
    const int M  = t & 15;
    const int kb = (t >> 4) * 2;
    const v2f ones = {1.f, 1.f};
    // Unconditional LDS loads (indices always in [0,16)), VALU select for the
    // strict-lower-triangular mask: avoids predicated ds_load around the WMMAs.
    float gv[8];
#pragma unroll
    for (int g = 0; g < 4; ++g) {
      gv[2 * g]     = gsum[4 * g + kb];
      gv[2 * g + 1] = gsum[4 * g + kb + 1];
    }
    v8f acc = {0.f, 0.f, 0.f, 0.f, 0.f, 0.f, 0.f, 0.f};
#pragma unroll
    for (int g = 0; g < 4; ++g) {     // D[i][*] = sum_{idx<i} gsum[idx]
      v2f a;
      a.x = ((4 * g + kb)     < M) ? gv[2 * g]     : 0.f;  // v_cndmask
      a.y = ((4 * g + kb + 1) < M) ? gv[2 * g + 1] : 0.f;
      acc = __builtin_amdgcn_wmma_f32_16x16x4_f32(false, a, false, ones,
                                                  (short)0, acc, false, false);
    }
    // D layout: vgpr v, lanes0-15 -> M=v; lanes16-31 -> M=v+8. Take column N=0.
    if (t == 0) {
#pragma unroll
      for (int v = 0; v < 8; ++v) goff[v] = acc[v];
    } else if (t == 16) {
#pragma unroll
      for (int v = 0; v < 8; ++v) goff[8 + v] = acc[v];
    }
  }
  __syncthreads();

  uint32_t off = (uint32_t)goff[t >> 6];
  for (int j = (t & ~63); j < t; ++j) off += part[j];
  uint32_t run = off;
  for (int i = 0; i < 128; ++i) {     // in-place exclusive rewrite of own chunk
    uint32_t v = hist[base + i];
    hist[base + i] = run;
    run += v;
  }
}

// -------------------------- K3: bin-grouped scatter ------------------------
__global__ __launch_bounds__(256) void scatter_kernel(const float* __restrict__ ch,
                                                      const uint32_t* __restrict__ base,
                                                      uint32_t* __restrict__ cursor,
                                                      uint32_t* __restrict__ rkey,
                                                      uint32_t* __restrict__ ridx) {
  uint32_t i = blockIdx.x * 256 + threadIdx.x;
  uint32_t k = fkey(ch[i]);
  uint32_t bin = k >> 15;
  uint32_t pos = base[bin] + atomicAdd(&cursor[bin], 1u);
  rkey[pos] = k;
  ridx[pos] = i;
}

// -------------------------- K5: passthrough copy ---------------------------
__global__ __launch_bounds__(256) void copy_kernel(const float* __restrict__ x,
                                                   float* __restrict__ out) {
  uint32_t i = blockIdx.x * 256 + threadIdx.x;
  out[i] = x[i];
}

// ---------------- TDM: 1-D tensor_load_to_lds of a bin's keys --------------
// D# per ISA ch.8: group0 = {count=1, lds_addr, global_addr(57b), type=2},
// group1 = {mask=0, data_size=4B, tensor_dim0=cnt, tensor_dim1=1,
//           tile_dim0=cnt, tile_dim1=1, dim0_stride=cnt}. Groups 2/3 zero.
__device__ __forceinline__ void tdm_load_rkey(uint32_t lds_addr, uint64_t gaddr, uint32_t cnt) {
  v4u g0;
  g0.x = 1u;                                            // descriptor count = 1
  g0.y = lds_addr;                                      // LDS byte address
  g0.z = (uint32_t)gaddr;                               // global_addr[31:0]
  g0.w = (uint32_t)((gaddr >> 32) & 0x01FFFFFFull) | 0x80000000u; // [56:32] | type=2
  v8i g1;
  g1[0] = (int)(2u << 16);                              // data_size = 4B
  g1[1] = (int)((cnt & 0xFFFFu) << 16);                 // tensor_dim0[15:0]
  g1[2] = (int)(((cnt >> 16) & 0xFFFFu) | (1u << 16));  // tensor_dim0[31:16], tensor_dim1=1
  g1[3] = (int)((cnt & 0xFFFFu) << 16);                 // tile_dim0 = cnt (<= 2048)
  g1[4] = 1;                                            // tile_dim1 = 1
  g1[5] = (int)cnt;                                     // tensor_dim0_stride
  g1[6] = 0;
  g1[7] = 0;
  v4i gz = {0, 0, 0, 0};
#if defined(__clang_major__) && (__clang_major__ >= 23)
  v8i gz8 = {0, 0, 0, 0, 0, 0, 0, 0};
  __builtin_amdgcn_tensor_load_to_lds(g0, g1, gz, gz, gz8, 0);
#else
  __builtin_amdgcn_tensor_load_to_lds(g0, g1, gz, gz, 0);
#endif
  __builtin_amdgcn_s_wait_tensorcnt(0);
}

// ------------- K4: per-bin low-15 LDS histogram -> exact ranks -------------
__global__ __launch_bounds__(256) void rank_kernel(const uint32_t* __restrict__ base,
                                                   const uint32_t* __restrict__ rkey,
                                                   const uint32_t* __restrict__ ridx,
                                                   const float* __restrict__ wgt,
                                                   const float* __restrict__ bia,
                                                   float* __restrict__ out) {
  extern __shared__ uint32_t smem[];
  uint32_t* h2    = smem;                 // 32768 counters (128 KB)
  uint32_t* stage = smem + L2Num;         // 2048-key TDM stage buffer
  uint32_t* part  = stage + StageCap;     // 256 scan partials

  const uint32_t bin   = blockIdx.x;
  const uint32_t start = base[bin];
  const uint32_t end   = (bin == NBins - 1) ? NPix : base[bin + 1];
  const uint32_t cnt   = end - start;
  if (cnt == 0) return;                   // uniform per block

  const int t = threadIdx.x;
  for (uint32_t i = t; i < L2Num; i += 256) h2[i] = 0u;

  const bool staged = (cnt <= StageCap);
  if (staged && t == 0) {                 // TDM issued once (EXEC-independent)
    uint32_t us = (uint32_t)__builtin_amdgcn_readfirstlane((int)start);
    uint32_t uc = (uint32_t)__builtin_amdgcn_readfirstlane((int)cnt);
    tdm_load_rkey((uint32_t)(uintptr_t)stage, (uint64_t)(uintptr_t)(rkey + us), uc);
  }
  __syncthreads();

  for (uint32_t r = t; r < cnt; r += 256) {
    uint32_t k = staged ? stage[r] : rkey[start + r];
    atomicAdd(&h2[k & 0x7FFFu], 1u);
  }
  __syncthreads();

  // exclusive scan of the 32768 LDS counters (256 threads x 128 chunks)
  {
    uint32_t s = 0;
    const int cb = t * (int)(L2Num / 256);
#pragma unroll 4
    for (int i = 0; i < (int)(L2Num / 256); ++i) s += h2[cb + i];
    part[t] = s;
    __syncthreads();
    uint32_t off = 0;
    for (int j = 0; j < t; ++j) off += part[j];
    uint32_t run = off;
    for (int i = 0; i < (int)(L2Num / 256); ++i) {
      uint32_t v = h2[cb + i];
      h2[cb + i] = run;
      run += v;
    }
    __syncthreads();
  }

  const float w0 = wgt[0], w1 = wgt[1], w2 = wgt[2];
  const float b0 = bia[0], b1 = bia[1], b2 = bia[2];
  for (uint32_t r = t; r < cnt; r += 256) {
    uint32_t k   = staged ? stage[r] : rkey[start + r];
    uint32_t idx = ridx[start + r];
    uint32_t row = idx >> 12, col = idx & 4095u;
    if (row == 0u || row == Hdim - 1 || col == 0u || col == Wdim - 1) continue;
    uint32_t low = k & 0x7FFFu;
    uint32_t lo  = start + h2[low];                                   // #(< v)
    uint32_t hi  = start + ((low == 0x7FFFu) ? cnt : h2[low + 1]);    // #(<= v)
    float e  = (float)lo;                       // all exact: <= 2^24
    float nf = (float)(hi - lo);
    float p  = (float)(NPix - hi);
    // match reference left-to-right association, then /3
    float val = (((((p * w0 + b0) + nf * w1) + b1) + e * w2) + b2) / 3.0f;
    out[NPix + idx] = val;                      // channel 1 interior
  }
}

// ---------------------------------------------------------------------------
extern "C" void kernel_launch(void* const* d_in, const int* in_sizes, int n_in,
                              void* d_out, int out_size, void* d_ws, size_t ws_size,
                              hipStream_t stream) {
  (void)in_sizes; (void)n_in; (void)out_size; (void)ws_size;
  const float* x = (const float*)d_in[0];
  const float* w = (const float*)d_in[1];
  const float* b = (const float*)d_in[2];
  float* out = (float*)d_out;

  // workspace layout: [hist 512KB][cursor 512KB][rkey 64MB][ridx 64MB]
  uint8_t*  ws     = (uint8_t*)d_ws;
  uint32_t* hist   = (uint32_t*)ws;
  uint32_t* cursor = hist + NBins;
  uint32_t* rkey   = (uint32_t*)(ws + (size_t)(1u << 20));
  uint32_t* ridx   = rkey + NPix;
  const float* ch1 = x + NPix;

  zero_kernel<<<(2 * NBins + 255) / 256, 256, 0, stream>>>(hist, 2 * NBins);
  hist_kernel<<<NPix / 256, 256, 0, stream>>>(ch1, hist);
  scan_kernel<<<1, 1024, 0, stream>>>(hist);                 // hist -> exclusive base
  scatter_kernel<<<NPix / 256, 256, 0, stream>>>(ch1, hist, cursor, rkey, ridx);
  copy_kernel<<<(3 * NPix) / 256, 256, 0, stream>>>(x, out); // passthrough; interior overwritten next
  rank_kernel<<<NBins, 256, RankSmemBytes, stream>>>(hist, rkey, ridx, w, b, out);
}